// DAS_FT_34462817583281
// MI455X (gfx1250) — compile-verified
//
#include <hip/hip_runtime.h>
#include <stdint.h>

#define TWO_PI_F    6.28318530718f
#define MIN_WIDTH_F 0.001f
#define MAX_NSAMP   2048
#define DAS_NBUF    8        // LDS row buffers (power of two)
#define DAS_DEPTH   7        // async prefetch depth, must be <= DAS_NBUF-1

#ifndef __has_builtin
#define __has_builtin(x) 0
#endif

// Use CDNA5 async global->LDS copies when the toolchain exposes them.
#if defined(__HIP_DEVICE_COMPILE__) && __has_builtin(__builtin_amdgcn_global_load_async_to_lds_b128)
#define DAS_ASYNC 1
#else
#define DAS_ASYNC 0
#endif

// Matches the builtin's parameter pointee type exactly:
//   __attribute__((__vector_size__(4 * sizeof(int)))) int
typedef int das_v4i __attribute__((vector_size(4 * sizeof(int))));

__device__ __forceinline__ void das_async_copy16(void* lds_dst, const void* gsrc) {
#if DAS_ASYNC
  // (global src, lds dst, imm offset, imm cpol). Pointer AS casts go through
  // uintptr_t (generic LDS addresses carry the LDS offset in the low 32 bits).
  __builtin_amdgcn_global_load_async_to_lds_b128(
      (__attribute__((address_space(1))) das_v4i*)(uintptr_t)gsrc,
      (__attribute__((address_space(3))) das_v4i*)(uint32_t)(uintptr_t)lds_dst,
      0, 0);
#else
  *reinterpret_cast<float4*>(lds_dst) = *reinterpret_cast<const float4*>(gsrc);
#endif
}

template <int N>
__device__ __forceinline__ void das_wait_async_le() {
#if DAS_ASYNC
#if __has_builtin(__builtin_amdgcn_s_wait_asynccnt)
  __builtin_amdgcn_s_wait_asynccnt(N);
#else
  asm volatile("s_wait_asynccnt %0" ::"i"(N) : "memory");
#endif
#endif
}

__global__ __launch_bounds__(1024, 1)
void das_kernel(const float* __restrict__ idata, const float* __restrict__ qdata,
                const float* __restrict__ grid,  const float* __restrict__ rx_ori,
                const float* __restrict__ ele_pos, const float* __restrict__ tstart,
                const float* __restrict__ pc, const float* __restrict__ pfs,
                const float* __restrict__ pfdemod, const float* __restrict__ pfnum,
                float* __restrict__ out,
                int nxmits, int nelems, int nsamps, int nx, int nz) {
  // DAS_NBUF-deep ring of I/Q sample rows for the streamed receive elements.
  // 2 * 8 * 8KB = 128 KB of the WGP's 320 KB LDS.
  __shared__ __align__(16) float bufI[DAS_NBUF][MAX_NSAMP];
  __shared__ __align__(16) float bufQ[DAS_NBUF][MAX_NSAMP];
  __shared__ float s_ex[256];

  const int x   = (int)blockIdx.x;
  const int tid = (int)threadIdx.x;
  const int z   = tid;                                   // one depth per thread
  const int dl  = (int)(((long long)x * (long long)nxmits) / (long long)nx);

  const float c      = pc[0];
  const float fs     = pfs[0];
  const float fdemod = pfdemod[0];
  const float fnum   = pfnum[0];
  const float ts     = tstart[dl];

  const int zc = (z < nz) ? z : (nz - 1);
  const size_t gb = ((size_t)x * (size_t)nz + (size_t)zc) * 3u;
  const float gx = grid[gb + 0];
  const float gy = grid[gb + 1];
  const float gz = grid[gb + 2];

  const float dxt = gx - rx_ori[x * 3 + 0];
  const float dyt = gy - rx_ori[x * 3 + 1];
  const float dzt = gz - rx_ori[x * 3 + 2];
  const float txdel = sqrtf(dxt * dxt + dyt * dyt + dzt * dzt);
  const float gyz2  = gy * gy + gz * gz;                 // for rxdel (ele y=z=0)

  const float* Irow0 = idata + (size_t)dl * (size_t)nelems * (size_t)nsamps;
  const float* Qrow0 = qdata + (size_t)dl * (size_t)nelems * (size_t)nsamps;
  const int qtr = nsamps >> 2;                           // threads per row (B128)

#define DAS_ISSUE(E, B)                                                           \
  do {                                                                            \
    if (tid < qtr) {                                                              \
      das_async_copy16(&bufI[(B)][tid * 4], Irow0 + (size_t)(E) * nsamps + tid * 4); \
    } else if (tid < 2 * qtr) {                                                   \
      int _t = tid - qtr;                                                         \
      das_async_copy16(&bufQ[(B)][_t * 4], Qrow0 + (size_t)(E) * nsamps + _t * 4);   \
    }                                                                             \
  } while (0)

  if (tid < nelems) s_ex[tid] = ele_pos[tid * 3];

  // Fill the software pipeline: DAS_DEPTH element rows in flight per wave.
#pragma unroll
  for (int p = 0; p < DAS_DEPTH; ++p) {
    if (p < nelems) DAS_ISSUE(p, p & (DAS_NBUF - 1));
  }
  __syncthreads();                                       // publish s_ex

  // Pass A: aperture count M (overlaps the first async loads' latency).
  // mask = (|gz/vx| >= fnum) | (|vx| <= MIN_WIDTH)  ==  (|gz| >= fnum*|vx|) | ...
  int M = 0;
  for (int e = 0; e < nelems; ++e) {
    const float vx = gx - s_ex[e];
    const bool mask = (fabsf(gz) >= fnum * fabsf(vx)) || (fabsf(vx) <= MIN_WIDTH_F);
    M += mask ? 1 : 0;
  }
  const bool  useHam = (M > 1);
  const float hamW   = TWO_PI_F / (float)(M > 1 ? M : 1);
  const float dscale = fs / c;                           // delays = (tx+rx)*fs/c - ts*fs
  const float dbias  = ts * fs;
  const float thA    = TWO_PI_F * fdemod / fs;           // theta = del*thA - thB
  const float thB    = TWO_PI_F * fdemod * (2.0f * gz / c);
  const bool  rot    = (fdemod != 0.0f);
  const float inv2pi = 0.15915494309189535f;

  float si = 0.0f, sq = 0.0f;
  int rank = 0;

  for (int e = 0; e < nelems; ++e) {
    // Async completion is in-order: <= DEPTH-1 outstanding => load(e) landed.
    if (e < nelems - (DAS_DEPTH - 1)) das_wait_async_le<DAS_DEPTH - 1>();
    else                              das_wait_async_le<0>();
    __syncthreads();                                     // load(e) visible to all waves;
                                                         // all waves done with compute(e-1)
    if (e + DAS_DEPTH < nelems)                          // stream element e+DEPTH into the
      DAS_ISSUE(e + DAS_DEPTH, (e + DAS_DEPTH) & (DAS_NBUF - 1)); // slot read at iter e-1

    const int   cur = e & (DAS_NBUF - 1);
    const float vx  = gx - s_ex[e];
    const bool mask = (fabsf(gz) >= fnum * fabsf(vx)) || (fabsf(vx) <= MIN_WIDTH_F);
    if (mask && z < nz) {
      const float apod = useHam ? (0.54f - 0.46f * __cosf(hamW * (float)rank)) : 1.0f;
      ++rank;

      const float rxdel = sqrtf(vx * vx + gyz2);
      const float del   = (txdel + rxdel) * dscale - dbias;
      const float s0f   = floorf(del);
      const float w     = del - s0f;
      const int   i0    = (int)s0f;
      const int   i1    = i0 + 1;
      const bool ok0 = (i0 >= 0) && (i0 < nsamps);
      const bool ok1 = (i1 >= 0) && (i1 < nsamps);
      const int  c0  = (i0 < 0) ? 0 : ((i0 >= nsamps) ? nsamps - 1 : i0);
      const int  c1  = (i1 < 0) ? 0 : ((i1 >= nsamps) ? nsamps - 1 : i1);

      float iv0 = bufI[cur][c0]; if (!ok0) iv0 = 0.0f;
      float iv1 = bufI[cur][c1]; if (!ok1) iv1 = 0.0f;
      float qv0 = bufQ[cur][c0]; if (!ok0) qv0 = 0.0f;
      float qv1 = bufQ[cur][c1]; if (!ok1) qv1 = 0.0f;

      const float om = 1.0f - w;
      float ifoc = iv0 * om + iv1 * w;
      float qfoc = qv0 * om + qv1 * w;

      if (rot) {
        float th = del * thA - thB;
        float tr = th * inv2pi;                          // range-reduce: HW sin valid
        tr -= rintf(tr);                                 // domain is limited
        float st, ct;
        __sincosf(tr * TWO_PI_F, &st, &ct);
        const float ir = ifoc * ct - qfoc * st;
        const float qr = qfoc * ct + ifoc * st;
        ifoc = ir; qfoc = qr;
      }
      si += ifoc * apod;
      sq += qfoc * apod;
    }
  }
#undef DAS_ISSUE

  if (z < nz) {
    out[(size_t)x * (size_t)nz + (size_t)z] = si;                             // idas
    out[(size_t)nx * (size_t)nz + (size_t)x * (size_t)nz + (size_t)z] = sq;   // qdas
  }
}

extern "C" void kernel_launch(void* const* d_in, const int* in_sizes, int n_in,
                              void* d_out, int out_size, void* d_ws, size_t ws_size,
                              hipStream_t stream) {
  (void)n_in; (void)d_ws; (void)ws_size; (void)out_size;
  const float* idata   = (const float*)d_in[0];
  const float* qdata   = (const float*)d_in[1];
  const float* grid    = (const float*)d_in[2];
  const float* rx_ori  = (const float*)d_in[3];
  const float* ele_pos = (const float*)d_in[4];
  const float* tstart  = (const float*)d_in[5];
  const float* pc      = (const float*)d_in[6];
  const float* pfs     = (const float*)d_in[7];
  const float* pfdemod = (const float*)d_in[8];
  const float* pfnum   = (const float*)d_in[9];
  float* out = (float*)d_out;

  const int nxmits = in_sizes[5];
  const int nelems = in_sizes[4] / 3;
  const int nsamps = in_sizes[0] / (nxmits * nelems);   // 2048
  const int nx     = in_sizes[3] / 3;                   // 128
  const int nz     = in_sizes[2] / (3 * nx);            // 1024

  dim3 blocks(nx);
  dim3 threads(nz);                                     // 1024 = 32 wave32
  das_kernel<<<blocks, threads, 0, stream>>>(idata, qdata, grid, rx_ori, ele_pos,
                                             tstart, pc, pfs, pfdemod, pfnum, out,
                                             nxmits, nelems, nsamps, nx, nz);
}